// ASTGCN_block_47064251629619
// MI455X (gfx1250) — compile-verified
//
#include <hip/hip_runtime.h>
#include <math.h>

// ASTGCN block for MI455X (gfx1250, wave32).
// Heavy GEMMs use V_WMMA_F32_16X16X4_F32 (native fp32 matrix path) with 1x4
// register blocking: one wave -> 16x64 output strip, one A fragment reused
// across 4 B fragments (4 WMMAs per K-step). global_prefetch_b8 lookahead on
// the streaming operand. tconv+rconv+ReLU+LayerNorm fused epilogue.

#define B_  32
#define N_  1024
#define C_  64
#define T_  24
#define K_  3
#define O_  64
#define FT_ 64
#define CT_ (C_ * T_)   // 1536

typedef __attribute__((ext_vector_type(2))) float v2f;
typedef __attribute__((ext_vector_type(8))) float v8f;

// D = A(16x4) * B(4x16) + C, fp32 (codegen-confirmed on gfx1250).
__device__ __forceinline__ v8f wmma4(v2f a, v2f b, v8f c) {
    return __builtin_amdgcn_wmma_f32_16x16x4_f32(
        false, a, false, b, (short)0, c, false, false);
}

// ---------------------------------------------------------------------------
// Small elementwise / reduction kernels (temporal + spatial attention setup)
// ---------------------------------------------------------------------------

// tlhs1[b,t,c] = sum_n x[b,n,c,t] * U1[n]
__global__ void k_tlhs1(const float* __restrict__ x, const float* __restrict__ U1,
                        float* __restrict__ o) {
    int idx = blockIdx.x * blockDim.x + threadIdx.x;
    if (idx >= B_ * T_ * C_) return;
    int c = idx % C_;
    int t = (idx / C_) % T_;
    int b = idx / (C_ * T_);
    float s = 0.f;
    for (int n = 0; n < N_; ++n)
        s += x[(((size_t)b * N_ + n) * C_ + c) * T_ + t] * U1[n];
    o[idx] = s;
}

// tlhs2[b,t,n] = sum_c tlhs1[b,t,c] * U2[c,n]
__global__ void k_tlhs2(const float* __restrict__ l1, const float* __restrict__ U2,
                        float* __restrict__ o) {
    int idx = blockIdx.x * blockDim.x + threadIdx.x;
    if (idx >= B_ * T_ * N_) return;
    int n  = idx % N_;
    int bt = idx / N_;
    float s = 0.f;
    for (int c = 0; c < C_; ++c)
        s += l1[(size_t)bt * C_ + c] * U2[(size_t)c * N_ + n];
    o[idx] = s;
}

// trhs[b,n,t] = sum_c U3[c] * x[b,n,c,t]
__global__ void k_trhs(const float* __restrict__ x, const float* __restrict__ U3,
                       float* __restrict__ o) {
    int idx = blockIdx.x * blockDim.x + threadIdx.x;
    if (idx >= B_ * N_ * T_) return;
    int t = idx % T_;
    size_t bn = idx / T_;
    float s = 0.f;
    for (int c = 0; c < C_; ++c)
        s += U3[c] * x[(bn * C_ + c) * T_ + t];
    o[idx] = s;
}

// tprod[b,s,t] = sum_n tlhs2[b,s,n] * trhs[b,n,t]
__global__ void k_tprod(const float* __restrict__ lhs, const float* __restrict__ rhs,
                        float* __restrict__ o) {
    int idx = blockIdx.x * blockDim.x + threadIdx.x;
    if (idx >= B_ * T_ * T_) return;
    int t = idx % T_;
    int s = (idx / T_) % T_;
    int b = idx / (T_ * T_);
    float acc = 0.f;
    for (int n = 0; n < N_; ++n)
        acc += lhs[((size_t)b * T_ + s) * N_ + n] * rhs[((size_t)b * N_ + n) * T_ + t];
    o[idx] = acc;
}

// E[b,s,y] = sum_t Ve[s,t] * sigmoid(tprod[b,t,y] + be[t,y])
__global__ void k_E(const float* __restrict__ tprod, const float* __restrict__ beT,
                    const float* __restrict__ Ve, float* __restrict__ o) {
    int idx = blockIdx.x * blockDim.x + threadIdx.x;
    if (idx >= B_ * T_ * T_) return;
    int y = idx % T_;
    int s = (idx / T_) % T_;
    int b = idx / (T_ * T_);
    float acc = 0.f;
    for (int t = 0; t < T_; ++t) {
        float z = tprod[((size_t)b * T_ + t) * T_ + y] + beT[t * T_ + y];
        acc += Ve[s * T_ + t] * (1.f / (1.f + expf(-z)));
    }
    o[idx] = acc;
}

// temporal_At = softmax over axis s, for each (b,y)
__global__ void k_tsoftmax(const float* __restrict__ E, float* __restrict__ o) {
    int idx = blockIdx.x * blockDim.x + threadIdx.x;
    if (idx >= B_ * T_) return;
    int y = idx % T_;
    int b = idx / T_;
    float mx = -3.4e38f;
    for (int s = 0; s < T_; ++s)
        mx = fmaxf(mx, E[((size_t)b * T_ + s) * T_ + y]);
    float sm = 0.f;
    for (int s = 0; s < T_; ++s)
        sm += expf(E[((size_t)b * T_ + s) * T_ + y] - mx);
    float inv = 1.f / sm;
    for (int s = 0; s < T_; ++s)
        o[((size_t)b * T_ + s) * T_ + y] =
            expf(E[((size_t)b * T_ + s) * T_ + y] - mx) * inv;
}

// x_TAt[b,n,c,s] = sum_t x[b,n,c,t] * tAt[b,t,s]
__global__ void k_xTAt(const float* __restrict__ x, const float* __restrict__ tAt,
                       float* __restrict__ o) {
    size_t idx = (size_t)blockIdx.x * blockDim.x + threadIdx.x;
    if (idx >= (size_t)B_ * N_ * C_ * T_) return;
    int s = (int)(idx % T_);
    size_t bnc = idx / T_;
    int b = (int)(bnc / ((size_t)N_ * C_));
    float acc = 0.f;
    for (int t = 0; t < T_; ++t)
        acc += x[bnc * T_ + t] * tAt[((size_t)b * T_ + t) * T_ + s];
    o[idx] = acc;
}

// slhs1[b,n,c] = sum_t xTAt[b,n,c,t] * W1[t]
__global__ void k_slhs1(const float* __restrict__ xT, const float* __restrict__ W1,
                        float* __restrict__ o) {
    size_t idx = (size_t)blockIdx.x * blockDim.x + threadIdx.x;
    if (idx >= (size_t)B_ * N_ * C_) return;
    float s = 0.f;
    for (int t = 0; t < T_; ++t) s += xT[idx * T_ + t] * W1[t];
    o[idx] = s;
}

// slhs2[b,n,t] = sum_c slhs1[b,n,c] * W2[c,t]
__global__ void k_slhs2(const float* __restrict__ l1, const float* __restrict__ W2,
                        float* __restrict__ o) {
    int idx = blockIdx.x * blockDim.x + threadIdx.x;
    if (idx >= B_ * N_ * T_) return;
    int t = idx % T_;
    size_t bn = idx / T_;
    float s = 0.f;
    for (int c = 0; c < C_; ++c) s += l1[bn * C_ + c] * W2[c * T_ + t];
    o[idx] = s;
}

// srhs[b,n,t] = sum_c W3[c] * xTAt[b,n,c,t]
__global__ void k_srhs(const float* __restrict__ xT, const float* __restrict__ W3,
                       float* __restrict__ o) {
    int idx = blockIdx.x * blockDim.x + threadIdx.x;
    if (idx >= B_ * N_ * T_) return;
    int t = idx % T_;
    size_t bn = idx / T_;
    float s = 0.f;
    for (int c = 0; c < C_; ++c) s += W3[c] * xT[(bn * C_ + c) * T_ + t];
    o[idx] = s;
}

// prod[idx] = sigmoid(prod[idx] + bs[n,y])  (in place)
__global__ void k_sig(float* __restrict__ prod, const float* __restrict__ bs) {
    size_t idx = (size_t)blockIdx.x * blockDim.x + threadIdx.x;
    if (idx >= (size_t)B_ * N_ * N_) return;
    float z = prod[idx] + bs[idx % ((size_t)N_ * N_)];
    prod[idx] = 1.f / (1.f + expf(-z));
}

// spatial_At = softmax(S, axis=m) in place, thread per (b,y) column
__global__ void k_csoftmax(float* __restrict__ S) {
    int idx = blockIdx.x * blockDim.x + threadIdx.x;
    if (idx >= B_ * N_) return;
    int y = idx % N_;
    int b = idx / N_;
    size_t base = (size_t)b * N_ * N_ + y;
    float mx = -3.4e38f;
    for (int m = 0; m < N_; ++m) mx = fmaxf(mx, S[base + (size_t)m * N_]);
    float sm = 0.f;
    for (int m = 0; m < N_; ++m) sm += expf(S[base + (size_t)m * N_] - mx);
    float inv = 1.f / sm;
    for (int m = 0; m < N_; ++m) {
        size_t a = base + (size_t)m * N_;
        S[a] = expf(S[a] - mx) * inv;
    }
}

// ---------------------------------------------------------------------------
// WMMA GEMM kernels. One wave -> 16x64 output strip (4 accumulators), 8
// waves/block; grids divide exactly so EXEC is always all-ones.
// ---------------------------------------------------------------------------

// prod[b,m,y] = sum_t slhs2[b,m,t] * srhs[b,y,t]   (K = 24)
__global__ void k_sprod(const float* __restrict__ lhs, const float* __restrict__ rhs,
                        float* __restrict__ prod) {
    int lane = threadIdx.x & 31;
    int tile = blockIdx.x * 8 + (threadIdx.x >> 5);
    const int TPB = 64 * 16;               // m-tiles x y-supertiles per batch
    int b   = tile / TPB;
    int r   = tile % TPB;
    int tm  = r >> 4, tyS = r & 15;
    int m  = tm * 16 + (lane & 15);
    int y0 = tyS * 64 + (lane & 15);
    int kh = (lane >> 4) << 1;
    const float* arow = lhs + ((size_t)b * N_ + m) * T_;
    const float* r0 = rhs + ((size_t)b * N_ + y0) * T_;
    v8f c0 = {}, c1 = {}, c2 = {}, c3 = {};
    for (int kb = 0; kb < T_; kb += 4) {
        int kk = kb + kh;
        v2f a  = *(const v2f*)(arow + kk);
        v2f b0 = *(const v2f*)(r0 + kk);
        v2f b1 = *(const v2f*)(r0 + (size_t)16 * T_ + kk);
        v2f b2 = *(const v2f*)(r0 + (size_t)32 * T_ + kk);
        v2f b3 = *(const v2f*)(r0 + (size_t)48 * T_ + kk);
        c0 = wmma4(a, b0, c0);
        c1 = wmma4(a, b1, c1);
        c2 = wmma4(a, b2, c2);
        c3 = wmma4(a, b3, c3);
    }
    int rbase = tm * 16 + ((lane >> 4) << 3);
    float* out = prod + (size_t)b * N_ * N_ + y0;
#pragma unroll
    for (int i = 0; i < 8; ++i) {
        float* orow = out + (size_t)(rbase + i) * N_;
        orow[0] = c0[i]; orow[16] = c1[i]; orow[32] = c2[i]; orow[48] = c3[i];
    }
}

// S[b,m,y] = sum_n Vs[m,n] * sig[b,n,y]   (K = 1024)
__global__ void k_vsmm(const float* __restrict__ Vs, const float* __restrict__ sig,
                       float* __restrict__ S) {
    int lane = threadIdx.x & 31;
    int tile = blockIdx.x * 8 + (threadIdx.x >> 5);
    const int TPB = 64 * 16;
    int b   = tile / TPB;
    int r   = tile % TPB;
    int tm  = r >> 4, tyS = r & 15;
    int m  = tm * 16 + (lane & 15);
    int y0 = tyS * 64 + (lane & 15);
    int kh = (lane >> 4) << 1;
    const float* arow = Vs + (size_t)m * N_;
    const float* sb   = sig + (size_t)b * N_ * N_ + y0;
    v8f c0 = {}, c1 = {}, c2 = {}, c3 = {};
    for (int kb = 0; kb < N_; kb += 4) {
        int kk = kb + kh;
        v2f a = *(const v2f*)(arow + kk);
        const float* p0 = sb + (size_t)kk * N_;
        const float* p1 = sb + (size_t)(kk + 1) * N_;
        v2f b0; b0.x = p0[0];  b0.y = p1[0];
        v2f b1; b1.x = p0[16]; b1.y = p1[16];
        v2f b2; b2.x = p0[32]; b2.y = p1[32];
        v2f b3; b3.x = p0[48]; b3.y = p1[48];
        // speculative lookahead on the streaming operand (global_prefetch_b8)
        if (kb + 32 < N_) __builtin_prefetch(p0 + (size_t)32 * N_, 0, 0);
        c0 = wmma4(a, b0, c0);
        c1 = wmma4(a, b1, c1);
        c2 = wmma4(a, b2, c2);
        c3 = wmma4(a, b3, c3);
    }
    int rbase = tm * 16 + ((lane >> 4) << 3);
    float* out = S + (size_t)b * N_ * N_ + y0;
#pragma unroll
    for (int i = 0; i < 8; ++i) {
        float* orow = out + (size_t)(rbase + i) * N_;
        orow[0] = c0[i]; orow[16] = c1[i]; orow[32] = c2[i]; orow[48] = c3[i];
    }
}

// rhs[b,n,ct] = sum_m (cheb_k[m,n] * At[b,m,n]) * x[b,m,ct]   (K = 1024)
// Hadamard (cheb ⊙ At) fused into the A-operand load.
__global__ void k_cheb(const float* __restrict__ chebk, const float* __restrict__ At,
                       const float* __restrict__ x, float* __restrict__ rhs) {
    int lane = threadIdx.x & 31;
    int tile = blockIdx.x * 8 + (threadIdx.x >> 5);
    const int TCS = CT_ / 64;              // 24 ct-supertiles
    const int TPB = 64 * TCS;
    int b   = tile / TPB;
    int r   = tile % TPB;
    int tn  = r / TCS, tcS = r % TCS;
    int n   = tn * 16 + (lane & 15);
    int ct0 = tcS * 64 + (lane & 15);
    int kh  = (lane >> 4) << 1;
    const float* Atb = At + (size_t)b * N_ * N_ + n;
    const float* chn = chebk + n;
    const float* xb  = x + (size_t)b * N_ * CT_ + ct0;
    v8f c0 = {}, c1 = {}, c2 = {}, c3 = {};
    for (int kb = 0; kb < N_; kb += 4) {
        int kk = kb + kh;
        v2f a;
        a.x = chn[(size_t)kk * N_] * Atb[(size_t)kk * N_];
        a.y = chn[(size_t)(kk + 1) * N_] * Atb[(size_t)(kk + 1) * N_];
        const float* p0 = xb + (size_t)kk * CT_;
        const float* p1 = xb + (size_t)(kk + 1) * CT_;
        v2f b0; b0.x = p0[0];  b0.y = p1[0];
        v2f b1; b1.x = p0[16]; b1.y = p1[16];
        v2f b2; b2.x = p0[32]; b2.y = p1[32];
        v2f b3; b3.x = p0[48]; b3.y = p1[48];
        if (kb + 32 < N_) {
            __builtin_prefetch(p0 + (size_t)32 * CT_, 0, 0);
            __builtin_prefetch(Atb + (size_t)(kk + 32) * N_, 0, 0);
        }
        c0 = wmma4(a, b0, c0);
        c1 = wmma4(a, b1, c1);
        c2 = wmma4(a, b2, c2);
        c3 = wmma4(a, b3, c3);
    }
    int rbase = tn * 16 + ((lane >> 4) << 3);
    float* out = rhs + (size_t)b * N_ * CT_ + ct0;
#pragma unroll
    for (int i = 0; i < 8; ++i) {
        float* orow = out + (size_t)(rbase + i) * CT_;
        orow[0] = c0[i]; orow[16] = c1[i]; orow[32] = c2[i]; orow[48] = c3[i];
    }
}

// acc[b,n,o,t] (+)= sum_c rhs[b,n,c,t] * Th[c,o]  (K = 64); one wave covers
// the full O=64 strip for a 16-row n-tile at fixed (b,t).
__global__ void k_theta(const float* __restrict__ rhs, const float* __restrict__ Th,
                        float* __restrict__ acc, int accum) {
    int lane = threadIdx.x & 31;
    int tile = blockIdx.x * 8 + (threadIdx.x >> 5);
    const int TPB = T_ * 64;               // t x n-tiles per batch
    int b  = tile / TPB;
    int r  = tile % TPB;
    int t  = r / 64;
    int tn = r % 64;
    int n  = tn * 16 + (lane & 15);
    int o0 = lane & 15;
    int kh = (lane >> 4) << 1;
    int rbase = tn * 16 + ((lane >> 4) << 3);
    v8f c0 = {}, c1 = {}, c2 = {}, c3 = {};
    if (accum) {
#pragma unroll
        for (int i = 0; i < 8; ++i) {
            const float* arow = acc + (((size_t)b * N_ + rbase + i) * O_ + o0) * T_ + t;
            c0[i] = arow[0];
            c1[i] = arow[(size_t)16 * T_];
            c2[i] = arow[(size_t)32 * T_];
            c3[i] = arow[(size_t)48 * T_];
        }
    }
    const float* ar = rhs + (((size_t)b * N_ + n) * C_) * T_ + t;
    for (int kb = 0; kb < C_; kb += 4) {
        int kk = kb + kh;
        v2f a;
        a.x = ar[(size_t)kk * T_];
        a.y = ar[(size_t)(kk + 1) * T_];
        const float* w0 = Th + (size_t)kk * O_ + o0;
        const float* w1 = Th + (size_t)(kk + 1) * O_ + o0;
        v2f b0; b0.x = w0[0];  b0.y = w1[0];
        v2f b1; b1.x = w0[16]; b1.y = w1[16];
        v2f b2; b2.x = w0[32]; b2.y = w1[32];
        v2f b3; b3.x = w0[48]; b3.y = w1[48];
        c0 = wmma4(a, b0, c0);
        c1 = wmma4(a, b1, c1);
        c2 = wmma4(a, b2, c2);
        c3 = wmma4(a, b3, c3);
    }
#pragma unroll
    for (int i = 0; i < 8; ++i) {
        float* orow = acc + (((size_t)b * N_ + rbase + i) * O_ + o0) * T_ + t;
        orow[0] = c0[i];
        orow[(size_t)16 * T_] = c1[i];
        orow[(size_t)32 * T_] = c2[i];
        orow[(size_t)48 * T_] = c3[i];
    }
}

// ---------------------------------------------------------------------------
// Fused epilogue: relu(acc) -> tconv(3-tap over T) + rconv(1x1) + relu + LN
// One 64-thread block per (b,n,t); thread f = output channel.
// ---------------------------------------------------------------------------
__global__ void k_out(const float* __restrict__ acc, const float* __restrict__ x,
                      const float* __restrict__ tw, const float* __restrict__ tb,
                      const float* __restrict__ rw, const float* __restrict__ rb,
                      const float* __restrict__ lng, const float* __restrict__ lnb,
                      float* __restrict__ out) {
    __shared__ float gcn[3 * O_];
    __shared__ float xr[C_];
    __shared__ float red[FT_];
    __shared__ float red2[FT_];
    int f = threadIdx.x;
    int pos = blockIdx.x;
    int t = pos % T_;
    int n = (pos / T_) % N_;
    int b = pos / (T_ * N_);

    size_t bx = ((size_t)b * N_ + n) * C_;
    xr[f] = x[(bx + f) * T_ + t];
    size_t bg = ((size_t)b * N_ + n) * O_;
    for (int dt = 0; dt < 3; ++dt) {
        int tt = t + dt - 1;
        float v = 0.f;
        if (tt >= 0 && tt < T_) {
            v = acc[(bg + f) * T_ + tt];
            v = v > 0.f ? v : 0.f;   // relu(spatial_gcn)
        }
        gcn[dt * O_ + f] = v;
    }
    __syncthreads();

    float sum = tb[f];
    for (int o = 0; o < O_; ++o) {
        const float* w = tw + ((size_t)f * O_ + o) * 3;
        sum += gcn[o] * w[0] + gcn[O_ + o] * w[1] + gcn[2 * O_ + o] * w[2];
    }
    float res = rb[f];
    for (int cc = 0; cc < C_; ++cc) res += xr[cc] * rw[(size_t)f * C_ + cc];
    float v = sum + res;
    v = v > 0.f ? v : 0.f;

    red[f] = v;
    red2[f] = v * v;
    __syncthreads();
    float mu = 0.f, m2 = 0.f;
    for (int j = 0; j < FT_; ++j) { mu += red[j]; m2 += red2[j]; }
    mu *= (1.f / FT_);
    m2 *= (1.f / FT_);
    float var = m2 - mu * mu;
    float yv = (v - mu) * rsqrtf(var + 1e-5f) * lng[f] + lnb[f];
    out[(((size_t)b * N_ + n) * FT_ + f) * T_ + t] = yv;
}

// ---------------------------------------------------------------------------

extern "C" void kernel_launch(void* const* d_in, const int* in_sizes, int n_in,
                              void* d_out, int out_size, void* d_ws, size_t ws_size,
                              hipStream_t stream) {
    const float* x     = (const float*)d_in[0];
    const float* W1    = (const float*)d_in[1];
    const float* W2    = (const float*)d_in[2];
    const float* W3    = (const float*)d_in[3];
    const float* bs    = (const float*)d_in[4];
    const float* Vs    = (const float*)d_in[5];
    const float* U1    = (const float*)d_in[6];
    const float* U2    = (const float*)d_in[7];
    const float* U3    = (const float*)d_in[8];
    const float* be    = (const float*)d_in[9];
    const float* Ve    = (const float*)d_in[10];
    const float* Theta = (const float*)d_in[11];
    const float* cheb  = (const float*)d_in[12];
    const float* tw    = (const float*)d_in[13];
    const float* tb    = (const float*)d_in[14];
    const float* rw    = (const float*)d_in[15];
    const float* rb    = (const float*)d_in[16];
    const float* lng   = (const float*)d_in[17];
    const float* lnb   = (const float*)d_in[18];
    // d_in[19] = time_strides == 1 (reference constant)
    float* out = (float*)d_out;
    float* ws  = (float*)d_ws;

    // Workspace layout (floats); ~693 MB total. x_TAt region is dead after
    // spatial attention and is reused as the Cheb GEMM output (same size).
    size_t off = 0;
    float* tlhs1 = ws + off; off += (size_t)B_ * T_ * C_;
    float* tlhs2 = ws + off; off += (size_t)B_ * T_ * N_;
    float* trhs  = ws + off; off += (size_t)B_ * N_ * T_;
    float* tprod = ws + off; off += (size_t)B_ * T_ * T_;
    float* Ebuf  = ws + off; off += (size_t)B_ * T_ * T_;
    float* tAt   = ws + off; off += (size_t)B_ * T_ * T_;
    float* xTAt  = ws + off; off += (size_t)B_ * N_ * C_ * T_;
    float* slhs1 = ws + off; off += (size_t)B_ * N_ * C_;
    float* slhs2 = ws + off; off += (size_t)B_ * N_ * T_;
    float* srhs  = ws + off; off += (size_t)B_ * N_ * T_;
    float* prod  = ws + off; off += (size_t)B_ * N_ * N_;
    float* Smat  = ws + off; off += (size_t)B_ * N_ * N_;
    float* accb  = ws + off; off += (size_t)B_ * N_ * O_ * T_;
    float* rhsb  = xTAt;  // alias

    const int TPB = 256;
    auto blocks = [](size_t n, int tpb) { return (unsigned)((n + tpb - 1) / tpb); };

    // --- temporal attention ---
    k_tlhs1<<<blocks((size_t)B_ * T_ * C_, TPB), TPB, 0, stream>>>(x, U1, tlhs1);
    k_tlhs2<<<blocks((size_t)B_ * T_ * N_, TPB), TPB, 0, stream>>>(tlhs1, U2, tlhs2);
    k_trhs<<<blocks((size_t)B_ * N_ * T_, TPB), TPB, 0, stream>>>(x, U3, trhs);
    k_tprod<<<blocks((size_t)B_ * T_ * T_, TPB), TPB, 0, stream>>>(tlhs2, trhs, tprod);
    k_E<<<blocks((size_t)B_ * T_ * T_, TPB), TPB, 0, stream>>>(tprod, be, Ve, Ebuf);
    k_tsoftmax<<<blocks((size_t)B_ * T_, TPB), TPB, 0, stream>>>(Ebuf, tAt);
    k_xTAt<<<blocks((size_t)B_ * N_ * C_ * T_, TPB), TPB, 0, stream>>>(x, tAt, xTAt);

    // --- spatial attention ---
    k_slhs1<<<blocks((size_t)B_ * N_ * C_, TPB), TPB, 0, stream>>>(xTAt, W1, slhs1);
    k_slhs2<<<blocks((size_t)B_ * N_ * T_, TPB), TPB, 0, stream>>>(slhs1, W2, slhs2);
    k_srhs<<<blocks((size_t)B_ * N_ * T_, TPB), TPB, 0, stream>>>(xTAt, W3, srhs);

    // WMMA: prod = lhs * rhs^T  (32 x [1024x1024, K=24]); 16x64 per wave
    k_sprod<<<(B_ * 64 * 16) / 8, 256, 0, stream>>>(slhs2, srhs, prod);
    k_sig<<<blocks((size_t)B_ * N_ * N_, TPB), TPB, 0, stream>>>(prod, bs);
    // WMMA: S = Vs * sigmoid(prod)  (32 x [1024x1024, K=1024]); 16x64 per wave
    k_vsmm<<<(B_ * 64 * 16) / 8, 256, 0, stream>>>(Vs, prod, Smat);
    k_csoftmax<<<blocks((size_t)B_ * N_, TPB), TPB, 0, stream>>>(Smat);

    // --- Chebyshev graph conv: 3x (WMMA K=1024 GEMM, then WMMA K=64 Theta) ---
    for (int k = 0; k < K_; ++k) {
        k_cheb<<<(B_ * 64 * (CT_ / 64)) / 8, 256, 0, stream>>>(
            cheb + (size_t)k * N_ * N_, Smat, x, rhsb);
        k_theta<<<(B_ * T_ * 64) / 8, 256, 0, stream>>>(
            rhsb, Theta + (size_t)k * C_ * O_, accb, k > 0 ? 1 : 0);
    }

    // --- fused tconv + rconv + relu + layernorm -> out (B,N,FT,T) ---
    k_out<<<(unsigned)((size_t)B_ * N_ * T_), FT_, 0, stream>>>(
        accb, x, tw, tb, rw, rb, lng, lnb, out);
}